// GCNLayer_75033078661648
// MI455X (gfx1250) — compile-verified
//
#include <hip/hip_runtime.h>
#include <hip/hip_bf16.h>

typedef __attribute__((ext_vector_type(2))) float v2f;
typedef __attribute__((ext_vector_type(8))) float v8f;

#define DIM 128   // d_in == d_out == 128

// ---------------------------------------------------------------------------
// Kernel 0: zero the h accumulator (ws is poisoned to 0xAA by the harness).
// ---------------------------------------------------------------------------
__global__ void gcn_zero_kernel(float4* __restrict__ p, int n4) {
    int i = blockIdx.x * blockDim.x + threadIdx.x;
    if (i < n4) p[i] = make_float4(0.f, 0.f, 0.f, 0.f);
}

// ---------------------------------------------------------------------------
// Kernel 1: edge scatter-add.  One wave32 per edge: src/dst are wave-uniform,
// lane l moves columns [4l, 4l+4) as one coalesced float4 gather + 4 f32
// atomics.  x and h are both L2-resident (5.12 MB each vs 192 MB L2).
// ---------------------------------------------------------------------------
__global__ void gcn_scatter_kernel(const float* __restrict__ x,
                                   const int*   __restrict__ src,
                                   const int*   __restrict__ dst,
                                   float*       __restrict__ h,
                                   int n_edges) {
    int gid  = blockIdx.x * blockDim.x + threadIdx.x;
    int e    = gid >> 5;            // one wave (32 lanes) per edge
    if (e >= n_edges) return;
    int lane = gid & 31;

    int s = src[e];                 // wave-uniform
    int d = dst[e];                 // wave-uniform

    const float4 v = *(const float4*)(x + (size_t)s * DIM + lane * 4);
    float* hp = h + (size_t)d * DIM + lane * 4;
    atomicAdd(hp + 0, v.x);
    atomicAdd(hp + 1, v.y);
    atomicAdd(hp + 2, v.z);
    atomicAdd(hp + 3, v.w);
}

// ---------------------------------------------------------------------------
// Kernel 2: out = relu(h @ W + b) via V_WMMA_F32_16X16X4_F32 (full fp32).
// One 16x16 output tile per wave; 8 waves/block cover a 16-row strip x 128
// cols.  K = 128 -> 32 WMMA ops per tile.
//
// Lane mapping (ISA 7.12.2, 32-bit operands, wave32):
//   half = lane>>4, l = lane&15
//   A 16x4 : row m = m0+l;      a = { h[m][kk+2*half], h[m][kk+1+2*half] }
//   B 4x16 : col n = n0+l;      b = { W[kk+2*half][n], W[kk+1+2*half][n] }
//   C 16x16: vgpr r -> (m0 + 8*half + r, n0 + l)
// ---------------------------------------------------------------------------
__global__ void gcn_gemm_relu_kernel(const float* __restrict__ h,
                                     const float* __restrict__ W,
                                     const float* __restrict__ b,
                                     float*       __restrict__ out) {
    const int lane = threadIdx.x & 31;
    const int wave = threadIdx.x >> 5;      // 0..7 : which 16-col tile
    const int m0   = blockIdx.x * 16;
    const int n0   = wave * 16;
    const int half = lane >> 4;             // 0 or 1
    const int l    = lane & 15;
    const int m    = m0 + l;                // A row held by this lane
    const int n    = n0 + l;                // B/C column held by this lane

    const float* __restrict__ hrow = h + (size_t)m * DIM;

    v8f acc = {};
    #pragma unroll
    for (int kk = 0; kk < DIM; kk += 4) {
        const int ka = kk + 2 * half;
        // A fragment: contiguous pair from h row  (K=ka, K=ka+1)
        v2f a = *(const v2f*)(hrow + ka);
        // B fragment: same K pair, column n of W
        v2f bb;
        bb.x = W[(size_t)ka * DIM + n];
        bb.y = W[(size_t)(ka + 1) * DIM + n];
        // D = A x B + C  (fp32, 16x16x4)
        acc = __builtin_amdgcn_wmma_f32_16x16x4_f32(
            /*neg_a=*/false, a, /*neg_b=*/false, bb,
            /*c_mod=*/(short)0, acc, /*reuse_a=*/false, /*reuse_b=*/false);
    }

    // Epilogue: bias + relu, store per C/D layout.
    const float bias  = b[n];
    const int   mbase = m0 + half * 8;
    #pragma unroll
    for (int r = 0; r < 8; ++r) {
        float v = acc[r] + bias;
        out[(size_t)(mbase + r) * DIM + n] = v > 0.f ? v : 0.f;
    }
}

// ---------------------------------------------------------------------------
// Launch:  inputs are (x, src, dst, W, b) per setup_inputs() order.
// ---------------------------------------------------------------------------
extern "C" void kernel_launch(void* const* d_in, const int* in_sizes, int n_in,
                              void* d_out, int out_size, void* d_ws, size_t ws_size,
                              hipStream_t stream) {
    const float* x   = (const float*)d_in[0];
    const int*   src = (const int*)  d_in[1];
    const int*   dst = (const int*)  d_in[2];
    const float* W   = (const float*)d_in[3];
    const float* b   = (const float*)d_in[4];
    float*       out = (float*)d_out;
    float*       h   = (float*)d_ws;          // [N, 128] f32 accumulator

    const int n_nodes = in_sizes[0] / DIM;    // 10000
    const int n_edges = in_sizes[1];          // 640000

    // 0) h = 0
    {
        int n4 = n_nodes * DIM / 4;
        int blk = 256;
        gcn_zero_kernel<<<(n4 + blk - 1) / blk, blk, 0, stream>>>((float4*)h, n4);
    }
    // 1) h[dst] += x[src]   (one wave per edge)
    {
        long long threads = (long long)n_edges * 32;
        int blk = 256;
        int grid = (int)((threads + blk - 1) / blk);
        gcn_scatter_kernel<<<grid, blk, 0, stream>>>(x, src, dst, h, n_edges);
    }
    // 2) out = relu(h @ W + b)   (625 blocks x 8 waves, one 16x16 tile/wave)
    {
        int grid = n_nodes / 16;              // 10000/16 = 625, exact
        gcn_gemm_relu_kernel<<<grid, 256, 0, stream>>>(h, W, b, out);
    }
}